// EGCN_71528385348100
// MI455X (gfx1250) — compile-verified
//
#include <hip/hip_runtime.h>

// ---------------------------------------------------------------------------
// GCN forward for MI455X (gfx1250, wave32).
//  * Dense GEMMs (X@W) use V_WMMA_F32_16X16X4_F32 (full fp32 accuracy).
//    Each wave computes a 16x64 output stripe (4 accumulators) so one A
//    fragment feeds 4 WMMAs. K and the fused-ReLU flag are template params
//    so the inner loop is branchless and fully unrollable.
//  * Edge aggregation / pooling use coalesced float4 + global_atomic_add_f32.
// Roofline: aggregation traffic (~5.2 GB) dominates at 23.3 TB/s; GEMMs are
// only 16.4 GFLOP so the f32 WMMA path is the right precision/perf tradeoff.
// ---------------------------------------------------------------------------

#define N_NODES  50000
#define N_EDGES  800000
#define N_GRAPHS 512
#define F_IN     128
#define HID      256
#define FC1_IN   (HID + 2)   // 258
#define FC1_OUT  196
#define DIM_OUT  16

typedef float v2f __attribute__((ext_vector_type(2)));
typedef float v8f __attribute__((ext_vector_type(8)));

// ---------------- degree / normalization ----------------

__global__ void k_init_deg(float* __restrict__ deg) {
    int n = blockIdx.x * blockDim.x + threadIdx.x;
    if (n < N_NODES) deg[n] = 1.0f;   // self-loop contribution
}

__global__ void k_edge_deg(const int* __restrict__ dst, float* __restrict__ deg) {
    int e = blockIdx.x * blockDim.x + threadIdx.x;
    if (e < N_EDGES) unsafeAtomicAdd(&deg[dst[e]], 1.0f);
}

__global__ void k_finish_dis(float* __restrict__ deg_to_dis) {
    int n = blockIdx.x * blockDim.x + threadIdx.x;
    if (n < N_NODES) {
        float d = deg_to_dis[n];
        deg_to_dis[n] = (d > 0.0f) ? rsqrtf(d) : 0.0f;
    }
}

// ---------------- WMMA f32 GEMM:  C[Nx256] = act(X[NxK]) @ W[Kx256] ---------
// One wave -> 16x64 stripe (4 x 16x16 tiles sharing the A fragment).
// Block = 128 thr (4 waves) -> one 16x256 row stripe. grid = N/16 blocks.
// N=50000 divides by 16 -> no guards -> EXEC all ones (WMMA requirement).

template <int K, bool RELU>
__global__ __launch_bounds__(128) void k_gemm_wmma(
    const float* __restrict__ X, const float* __restrict__ W,
    float* __restrict__ C)
{
    const int lane   = threadIdx.x & 31;
    const int wave   = threadIdx.x >> 5;
    const int laneLo = lane & 15;
    const int kHi    = (lane >> 4) << 1;            // 0 or 2 (K half-split)
    const int row    = blockIdx.x * 16 + laneLo;    // A: M = laneLo
    const int col    = wave * 64 + laneLo;          // B/D: N = laneLo (+16*t)

    const float* xrow = X + (size_t)row * K;
    v8f acc0 = {}, acc1 = {}, acc2 = {}, acc3 = {};

    for (int k0 = 0; k0 < K; k0 += 4) {
        const int ka = k0 + kHi;
        float a0 = xrow[ka];
        float a1 = xrow[ka + 1];
        if (RELU) { a0 = fmaxf(a0, 0.0f); a1 = fmaxf(a1, 0.0f); }
        v2f a; a.x = a0; a.y = a1;

        const float* w0 = W + (size_t)ka * HID + col;
        const float* w1 = W + (size_t)(ka + 1) * HID + col;

        v2f b0; b0.x = w0[0];  b0.y = w1[0];
        v2f b1; b1.x = w0[16]; b1.y = w1[16];
        v2f b2; b2.x = w0[32]; b2.y = w1[32];
        v2f b3; b3.x = w0[48]; b3.y = w1[48];

        acc0 = __builtin_amdgcn_wmma_f32_16x16x4_f32(false, a, false, b0, (short)0, acc0, false, false);
        acc1 = __builtin_amdgcn_wmma_f32_16x16x4_f32(false, a, false, b1, (short)0, acc1, false, false);
        acc2 = __builtin_amdgcn_wmma_f32_16x16x4_f32(false, a, false, b2, (short)0, acc2, false, false);
        acc3 = __builtin_amdgcn_wmma_f32_16x16x4_f32(false, a, false, b3, (short)0, acc3, false, false);
    }

    // D layout: VGPR r -> M = r (lanes 0-15) / 8+r (lanes 16-31), N = laneLo
    const int mBase = blockIdx.x * 16 + ((lane >> 4) << 3);
    float* cbase = C + (size_t)mBase * HID + col;
#pragma unroll
    for (int r = 0; r < 8; ++r) {
        float* crow = cbase + (size_t)r * HID;
        crow[0]  = acc0[r];
        crow[16] = acc1[r];
        crow[32] = acc2[r];
        crow[48] = acc3[r];
    }
}

// ---------------- aggregation -----------------------------------------------
// out[n][f] = dis[n]^2 * xw[n][f] + bias[f]   (self loop + bias init)

__global__ void k_init_out(const float4* __restrict__ xw,
                           const float* __restrict__ dis,
                           const float* __restrict__ bias,
                           float4* __restrict__ out)
{
    int idx = blockIdx.x * blockDim.x + threadIdx.x;   // N_NODES*64 threads
    if (idx >= N_NODES * (HID / 4)) return;
    int n = idx >> 6;
    int c = idx & 63;
    float s  = dis[n];
    float nn = s * s;
    float4 v = xw[idx];
    const float4 b = ((const float4*)bias)[c];
    float4 o;
    o.x = nn * v.x + b.x;  o.y = nn * v.y + b.y;
    o.z = nn * v.z + b.z;  o.w = nn * v.w + b.w;
    out[idx] = o;
}

// out[dst] += dis[src]*dis[dst] * xw[src]   (one thread = one edge x 4 feats)
__global__ void k_scatter(const float4* __restrict__ xw,
                          const float* __restrict__ dis,
                          const int* __restrict__ src,
                          const int* __restrict__ dst,
                          float* __restrict__ out)
{
    long long tid = (long long)blockIdx.x * blockDim.x + threadIdx.x;
    int e = (int)(tid >> 6);
    int c = (int)(tid & 63);
    if (e >= N_EDGES) return;
    int s = src[e], d = dst[e];
    float norm = dis[s] * dis[d];
    float4 v = xw[(size_t)s * 64 + c];
    float* o = out + (size_t)d * HID + c * 4;
    unsafeAtomicAdd(o + 0, norm * v.x);
    unsafeAtomicAdd(o + 1, norm * v.y);
    unsafeAtomicAdd(o + 2, norm * v.z);
    unsafeAtomicAdd(o + 3, norm * v.w);
}

// ---------------- pooling + head --------------------------------------------

__global__ void k_zero(float* __restrict__ p, int n) {
    int i = blockIdx.x * blockDim.x + threadIdx.x;
    if (i < n) p[i] = 0.0f;
}

__global__ void k_pool(const float4* __restrict__ h,
                       const int* __restrict__ batch,
                       float* __restrict__ sums)
{
    int idx = blockIdx.x * blockDim.x + threadIdx.x;   // N_NODES*64
    if (idx >= N_NODES * (HID / 4)) return;
    int n = idx >> 6;
    int c = idx & 63;
    float4 v = h[idx];                                  // apply ReLU of conv3
    v.x = fmaxf(v.x, 0.0f); v.y = fmaxf(v.y, 0.0f);
    v.z = fmaxf(v.z, 0.0f); v.w = fmaxf(v.w, 0.0f);
    int g = batch[n];
    float* o = sums + (size_t)g * HID + c * 4;
    unsafeAtomicAdd(o + 0, v.x);
    unsafeAtomicAdd(o + 1, v.y);
    unsafeAtomicAdd(o + 2, v.z);
    unsafeAtomicAdd(o + 3, v.w);
}

__global__ void k_counts(const int* __restrict__ batch, float* __restrict__ cnt) {
    int n = blockIdx.x * blockDim.x + threadIdx.x;
    if (n < N_NODES) unsafeAtomicAdd(&cnt[batch[n]], 1.0f);
}

__global__ void k_hg(const float* __restrict__ sums, const float* __restrict__ cnt,
                     const float* __restrict__ mol, const float* __restrict__ rings,
                     float* __restrict__ hg)
{
    int idx = blockIdx.x * blockDim.x + threadIdx.x;   // G * 258
    if (idx >= N_GRAPHS * FC1_IN) return;
    int g = idx / FC1_IN;
    int f = idx - g * FC1_IN;
    float v;
    if (f < HID)       v = sums[(size_t)g * HID + f] / fmaxf(cnt[g], 1.0f);
    else if (f == HID) v = mol[g];
    else               v = rings[g];
    hg[idx] = v;
}

__global__ void k_fc1(const float* __restrict__ hg, const float* __restrict__ W,
                      const float* __restrict__ b, float* __restrict__ f1)
{
    int idx = blockIdx.x * blockDim.x + threadIdx.x;   // G * 196
    if (idx >= N_GRAPHS * FC1_OUT) return;
    int g = idx / FC1_OUT;
    int o = idx - g * FC1_OUT;
    float acc = b[o];
    const float* row = hg + (size_t)g * FC1_IN;
    for (int k = 0; k < FC1_IN; ++k)
        acc = fmaf(row[k], W[(size_t)k * FC1_OUT + o], acc);
    f1[idx] = fmaxf(acc, 0.0f);
}

__global__ void k_fc2(const float* __restrict__ f1, const float* __restrict__ W,
                      const float* __restrict__ b, float* __restrict__ out)
{
    int idx = blockIdx.x * blockDim.x + threadIdx.x;   // G * 16
    if (idx >= N_GRAPHS * DIM_OUT) return;
    int g = idx >> 4;
    int o = idx & 15;
    float acc = b[o];
    const float* row = f1 + (size_t)g * FC1_OUT;
    for (int k = 0; k < FC1_OUT; ++k)
        acc = fmaf(row[k], W[(size_t)k * DIM_OUT + o], acc);
    out[idx] = acc;
}

// ---------------- launcher --------------------------------------------------

extern "C" void kernel_launch(void* const* d_in, const int* in_sizes, int n_in,
                              void* d_out, int out_size, void* d_ws, size_t ws_size,
                              hipStream_t stream) {
    const float* x      = (const float*)d_in[0];
    const int*   ei     = (const int*)d_in[1];
    const int*   batch  = (const int*)d_in[2];
    const float* mol    = (const float*)d_in[3];
    const float* rings  = (const float*)d_in[4];
    const float* W1 = (const float*)d_in[5],  *b1 = (const float*)d_in[6];
    const float* W2 = (const float*)d_in[7],  *b2 = (const float*)d_in[8];
    const float* W3 = (const float*)d_in[9],  *b3 = (const float*)d_in[10];
    const float* fcW1 = (const float*)d_in[11], *fcb1 = (const float*)d_in[12];
    const float* fcW2 = (const float*)d_in[13], *fcb2 = (const float*)d_in[14];
    const int* srcp = ei;
    const int* dstp = ei + N_EDGES;
    float* outp = (float*)d_out;

    // workspace carve-up (floats)
    float* ws = (float*)d_ws;
    size_t off = 0;
    float* bufA = ws + off; off += (size_t)N_NODES * HID;   // GEMM outputs
    float* bufB = ws + off; off += (size_t)N_NODES * HID;   // aggregation outputs
    float* dis  = ws + off; off += N_NODES;                 // deg -> deg^-1/2
    float* sums = ws + off; off += (size_t)N_GRAPHS * HID;  // pooled sums
    float* cnt  = ws + off; off += N_GRAPHS;                // graph node counts
    float* hg   = ws + off; off += (size_t)N_GRAPHS * FC1_IN;
    float* f1   = ws + off; off += (size_t)N_GRAPHS * FC1_OUT;

    const int T = 256;
    const int nBlkN   = (N_NODES + T - 1) / T;          // 196
    const int nBlkE   = (N_EDGES + T - 1) / T;          // 3125
    const int nBlkNF  = N_NODES * (HID / 4) / T;        // 12500
    const int nBlkEF  = N_EDGES * (HID / 4) / T;        // 200000
    const int gemmGrid = N_NODES / 16;                  // 3125 blocks x 128 thr

    // degree normalization
    k_init_deg  <<<nBlkN, T, 0, stream>>>(dis);
    k_edge_deg  <<<nBlkE, T, 0, stream>>>(dstp, dis);
    k_finish_dis<<<nBlkN, T, 0, stream>>>(dis);

    // ---- layer 1: xw = X @ W1 ; out = norm-agg(xw) + b1
    k_gemm_wmma<F_IN, false><<<gemmGrid, 128, 0, stream>>>(x, W1, bufA);
    k_init_out <<<nBlkNF, T, 0, stream>>>((const float4*)bufA, dis, b1, (float4*)bufB);
    k_scatter  <<<nBlkEF, T, 0, stream>>>((const float4*)bufA, dis, srcp, dstp, bufB);

    // ---- layer 2 (ReLU fused into GEMM A-load)
    k_gemm_wmma<HID, true><<<gemmGrid, 128, 0, stream>>>(bufB, W2, bufA);
    k_init_out <<<nBlkNF, T, 0, stream>>>((const float4*)bufA, dis, b2, (float4*)bufB);
    k_scatter  <<<nBlkEF, T, 0, stream>>>((const float4*)bufA, dis, srcp, dstp, bufB);

    // ---- layer 3
    k_gemm_wmma<HID, true><<<gemmGrid, 128, 0, stream>>>(bufB, W3, bufA);
    k_init_out <<<nBlkNF, T, 0, stream>>>((const float4*)bufA, dis, b3, (float4*)bufB);
    k_scatter  <<<nBlkEF, T, 0, stream>>>((const float4*)bufA, dis, srcp, dstp, bufB);

    // ---- global mean pool (ReLU of conv3 fused into pooling read)
    k_zero  <<<(N_GRAPHS * HID + N_GRAPHS + T - 1) / T, T, 0, stream>>>(sums, N_GRAPHS * HID + N_GRAPHS);
    k_pool  <<<nBlkNF, T, 0, stream>>>((const float4*)bufB, batch, sums);
    k_counts<<<nBlkN,  T, 0, stream>>>(batch, cnt);
    k_hg    <<<(N_GRAPHS * FC1_IN + T - 1) / T, T, 0, stream>>>(sums, cnt, mol, rings, hg);

    // ---- MLP head
    k_fc1<<<(N_GRAPHS * FC1_OUT + T - 1) / T, T, 0, stream>>>(hg, fcW1, fcb1, f1);
    k_fc2<<<(N_GRAPHS * DIM_OUT + T - 1) / T, T, 0, stream>>>(f1, fcW2, fcb2, outp);
}